// GCN_62732292326150
// MI455X (gfx1250) — compile-verified
//
#include <hip/hip_runtime.h>
#include <hip/hip_bf16.h>

typedef __attribute__((ext_vector_type(2))) float v2f;
typedef __attribute__((ext_vector_type(8))) float v8f;

#define D_CH 128

// ---------------------------------------------------------------------------
// helpers
// ---------------------------------------------------------------------------
__device__ __forceinline__ void atomic_add_f32(float* p, float v) {
#if defined(__HIP_DEVICE_COMPILE__)
    unsafeAtomicAdd(p, v);   // lowers to global_atomic_add_f32 on gfx1250
#else
    atomicAdd(p, v);
#endif
}

// ---------------------------------------------------------------------------
// fill a float buffer with a constant
// ---------------------------------------------------------------------------
__global__ void fill_f32(float* __restrict__ p, float v, int n) {
    int i = blockIdx.x * blockDim.x + threadIdx.x;
    if (i < n) p[i] = v;
}

// deg[dst] += w   (deg pre-filled with 1.0 for the self-loop)
__global__ void deg_accum(float* __restrict__ deg, const int* __restrict__ dst,
                          const float* __restrict__ w, int E) {
    int e = blockIdx.x * blockDim.x + threadIdx.x;
    if (e < E) atomic_add_f32(&deg[dst[e]], w[e]);
}

// deg -> rsqrt(deg) in place
__global__ void rsqrt_inplace(float* __restrict__ d, int n) {
    int i = blockIdx.x * blockDim.x + threadIdx.x;
    if (i < n) d[i] = __frsqrt_rn(d[i]);
}

// norm[e] = dinv[src]*w*dinv[dst]
__global__ void edge_norm(float* __restrict__ norm, const int* __restrict__ src,
                          const int* __restrict__ dst, const float* __restrict__ w,
                          const float* __restrict__ dinv, int E) {
    int e = blockIdx.x * blockDim.x + threadIdx.x;
    if (e < E) norm[e] = dinv[src[e]] * w[e] * dinv[dst[e]];
}

// ---------------------------------------------------------------------------
// H = X @ W     X:[nRows x 128] row-major, W:[128 x 128] row-major
// One wave (32 lanes) per 16-row M tile; 8 N-tiles of 16; K=128 as 32 x K4
// V_WMMA_F32_16X16X4_F32 lane layouts per ISA 7.12.2:
//   A 16x4 : lanes 0-15 -> K=0,1 ; lanes 16-31 -> K=2,3 ; row = lane&15
//   B 4x16 : lanes 0-15 -> K=0,1 ; lanes 16-31 -> K=2,3 ; col = lane&15
//   C/D    : VGPR r = row (lane<16 ? r : r+8), col = lane&15
// ---------------------------------------------------------------------------
__global__ __launch_bounds__(32) void gemm_wmma_f32(
        const float* __restrict__ X, const float* __restrict__ W,
        float* __restrict__ H) {
#if defined(__HIP_DEVICE_COMPILE__)
    const int lane  = threadIdx.x;
    const int m0    = blockIdx.x * 16;
    const int rowA  = m0 + (lane & 15);
    const int koff  = (lane >> 4) * 2;      // 0 or 2
    const int colB  = lane & 15;

    v8f acc[8];
#pragma unroll
    for (int n = 0; n < 8; ++n) acc[n] = v8f{};

    const float* __restrict__ xrow = X + (size_t)rowA * D_CH;

    for (int ks = 0; ks < 32; ++ks) {
        const int kb = ks * 4;
        v2f a;
        a.x = xrow[kb + koff];
        a.y = xrow[kb + koff + 1];
        const float* __restrict__ w0 = W + (size_t)(kb + koff) * D_CH;
#pragma unroll
        for (int n = 0; n < 8; ++n) {
            v2f b;
            b.x = w0[n * 16 + colB];
            b.y = w0[D_CH + n * 16 + colB];
            acc[n] = __builtin_amdgcn_wmma_f32_16x16x4_f32(
                false, a, false, b, (short)0, acc[n], false, false);
        }
    }

    const int outRow  = m0 + ((lane >> 4) * 8);
    const int outCol0 = lane & 15;
#pragma unroll
    for (int n = 0; n < 8; ++n) {
#pragma unroll
        for (int r = 0; r < 8; ++r) {
            H[(size_t)(outRow + r) * D_CH + n * 16 + outCol0] = acc[n][r];
        }
    }
#endif
}

// ---------------------------------------------------------------------------
// agg[dst] += norm[e] * h[src]   one wave per edge, 4 channels per lane
// ---------------------------------------------------------------------------
__global__ __launch_bounds__(256) void edge_scatter(
        const float* __restrict__ h, const float* __restrict__ norm,
        const int* __restrict__ src, const int* __restrict__ dst,
        float* __restrict__ agg, int E) {
    const int wave = blockIdx.x * (blockDim.x / 32) + (threadIdx.x >> 5);
    const int lane = threadIdx.x & 31;
    if (wave >= E) return;
    const int   s  = src[wave];
    const int   d  = dst[wave];
    const float nw = norm[wave];
    const float* __restrict__ hs = h + (size_t)s * D_CH;
    float* __restrict__ ad = agg + (size_t)d * D_CH;
#pragma unroll
    for (int j = 0; j < 4; ++j) {
        const int c = lane + j * 32;
        atomic_add_f32(&ad[c], nw * hs[c]);
    }
}

// ---------------------------------------------------------------------------
// xout = relu(agg + h * dinv^2 + b)
// ---------------------------------------------------------------------------
__global__ void finish_layer(const float* __restrict__ agg,
                             const float* __restrict__ h,
                             const float* __restrict__ dinv,
                             const float* __restrict__ b,
                             float* __restrict__ xout, int total) {
    int i = blockIdx.x * blockDim.x + threadIdx.x;
    if (i >= total) return;
    const int node = i >> 7;
    const int ch   = i & (D_CH - 1);
    const float di = dinv[node];
    const float v  = agg[i] + h[i] * di * di + b[ch];
    xout[i] = fmaxf(v, 0.0f);
}

// ---------------------------------------------------------------------------
// out[b] = (x1[idx] + x2[idx] + x3[idx]) / 3   (idx == N -> zero row)
// ---------------------------------------------------------------------------
__global__ void lookup_mean(const float* __restrict__ x1,
                            const float* __restrict__ x2,
                            const float* __restrict__ x3,
                            const int* __restrict__ idx,
                            float* __restrict__ out, int nNodes, int total) {
    int i = blockIdx.x * blockDim.x + threadIdx.x;
    if (i >= total) return;
    const int bi = i >> 7;
    const int ch = i & (D_CH - 1);
    const int id = idx[bi];
    float v = 0.0f;
    if (id < nNodes) {
        const size_t o = (size_t)id * D_CH + ch;
        v = (x1[o] + x2[o] + x3[o]) * (1.0f / 3.0f);
    }
    out[i] = v;
}

// ---------------------------------------------------------------------------
extern "C" void kernel_launch(void* const* d_in, const int* in_sizes, int n_in,
                              void* d_out, int out_size, void* d_ws, size_t ws_size,
                              hipStream_t stream) {
    const int*   inputs_idx  = (const int*)  d_in[0];
    const float* x           = (const float*)d_in[1];
    const int*   edge_index  = (const int*)  d_in[2];
    const float* edge_weight = (const float*)d_in[3];
    const float* Wm[3] = { (const float*)d_in[4], (const float*)d_in[6], (const float*)d_in[8] };
    const float* bv[3] = { (const float*)d_in[5], (const float*)d_in[7], (const float*)d_in[9] };

    const int B = in_sizes[0];
    const int N = in_sizes[1] / D_CH;
    const int E = in_sizes[2] / 2;
    (void)B; (void)n_in; (void)ws_size;
    const int* src = edge_index;
    const int* dst = edge_index + E;

    // workspace layout (floats)
    float* ws   = (float*)d_ws;
    const size_t nd = (size_t)N * D_CH;
    float* dinv = ws;                 // N        (holds deg, then rsqrt(deg))
    float* norm = dinv + N;           // E
    float* h    = norm + E;           // N*D
    float* agg  = h + nd;             // N*D
    float* xl[3];
    xl[0] = agg + nd;                 // N*D each
    xl[1] = xl[0] + nd;
    xl[2] = xl[1] + nd;

    const int TB = 256;
    const int mTiles   = N / 16;                          // 3125
    const int edgeBlk  = (E + TB - 1) / TB;
    const int nodeBlk  = (N + TB - 1) / TB;
    const int ndBlk    = (int)((nd + TB - 1) / TB);
    const int waveBlk  = (E + (TB / 32) - 1) / (TB / 32); // 1 wave per edge
    const int outBlk   = (out_size + TB - 1) / TB;

    // ---- graph preprocessing (shared across the 3 layers) ----
    fill_f32<<<nodeBlk, TB, 0, stream>>>(dinv, 1.0f, N);          // self-loop
    deg_accum<<<edgeBlk, TB, 0, stream>>>(dinv, dst, edge_weight, E);
    rsqrt_inplace<<<nodeBlk, TB, 0, stream>>>(dinv, N);
    edge_norm<<<edgeBlk, TB, 0, stream>>>(norm, src, dst, edge_weight, dinv, E);

    // ---- three GCN layers ----
    const float* xin = x;
    for (int l = 0; l < 3; ++l) {
        fill_f32<<<ndBlk, TB, 0, stream>>>(agg, 0.0f, (int)nd);
        gemm_wmma_f32<<<mTiles, 32, 0, stream>>>(xin, Wm[l], h);
        edge_scatter<<<waveBlk, TB, 0, stream>>>(h, norm, src, dst, agg, E);
        finish_layer<<<ndBlk, TB, 0, stream>>>(agg, h, dinv, bv[l], xl[l], (int)nd);
        xin = xl[l];
    }

    // ---- embedding lookup + mean of the three layers ----
    lookup_mean<<<outBlk, TB, 0, stream>>>(xl[0], xl[1], xl[2], inputs_idx,
                                           (float*)d_out, N, out_size);
}